// GlobalPool_59365037965410
// MI455X (gfx1250) — compile-verified
//
#include <hip/hip_runtime.h>
#include <math.h>

// Problem constants (fixed by the reference).
#define FDIM   256
#define NNODES 204800
#define BGRAPH 4096
#define CH     24      // node rows staged in LDS per chunk (double-buffered)
#define SS     260     // padded LDS row stride (floats): 260*4B % 16 == 0, 260 % 64 == 4

typedef __attribute__((ext_vector_type(2))) float v2f;
typedef __attribute__((ext_vector_type(8))) float v8f;

__device__ __forceinline__ float warp_sum(float v) {
#pragma unroll
  for (int off = 16; off > 0; off >>= 1) v += __shfl_xor(v, off, 32);
  return v;
}

__device__ __forceinline__ v8f wmma_f32(v2f a, v2f b, v8f c) {
  // D = A(16x4 f32) * B(4x16 f32) + C(16x16 f32), wave32
  return __builtin_amdgcn_wmma_f32_16x16x4_f32(
      /*neg_a=*/false, a, /*neg_b=*/false, b,
      /*c_mod=*/(short)0, c, /*reuse_a=*/false, /*reuse_b=*/false);
}

#if __has_builtin(__builtin_amdgcn_tensor_load_to_lds) && \
    __has_builtin(__builtin_amdgcn_s_wait_tensorcnt) &&   \
    __has_builtin(__builtin_amdgcn_readfirstlane)
#define USE_TDM 1
#endif

#ifdef USE_TDM
typedef unsigned int v4u __attribute__((ext_vector_type(4)));
typedef int v8i __attribute__((ext_vector_type(8)));
typedef int v4i __attribute__((ext_vector_type(4)));

// Issue one TDM 2D-tile load: rows x 256 f32, row-major (stride 256 elems),
// into LDS at lds_addr with hardware padding of 4 DWORDs every 256 DWORDs
// (i.e. LDS row stride becomes SS=260 floats, bank-conflict-free layout).
__device__ __forceinline__ void tdm_load_rows(unsigned lds_addr,
                                              const float* gsrc, int rows) {
  const unsigned long long ga = (unsigned long long)(uintptr_t)gsrc;
  v4u g0;
  g0[0] = 1u;                                   // count=1, user descriptor
  g0[1] = lds_addr;                             // lds_addr[31:0]
  g0[2] = (unsigned)ga;                         // global_addr[31:0]
  g0[3] = (unsigned)((ga >> 32) & 0x01FFFFFFu)  // global_addr[56:32]
          | (2u << 30);                         // type = 2 ("image")
  v8i g1;
  g1[0] = (int)((2u << 16)      // data_size = 4 bytes
              | (1u << 20)      // pad_enable
              | (7u << 22)      // pad_interval code 7 -> every 256 DWORDs
              | (3u << 25));    // pad_amount  code 3 -> 4 DWORDs
  g1[1] = (int)(256u << 16);    // tensor_dim0[15:0] = 256 (bits 79:48)
  g1[2] = (int)((unsigned)rows << 16); // tensor_dim0 hi = 0; tensor_dim1 lo = rows
  g1[3] = (int)(256u << 16);    // tensor_dim1 hi = 0; tile_dim0 = 256
  g1[4] = rows;                 // tile_dim1 = rows; tile_dim2 = 0
  g1[5] = 256;                  // tensor_dim0_stride[31:0] = 256 elems
  g1[6] = 0;                    // stride hi bits / tensor_dim1_stride lo
  g1[7] = 0;
  const v4i z4 = {0, 0, 0, 0};           // groups 2/3 unused (2D tensor)
  const v8i z8 = {0, 0, 0, 0, 0, 0, 0, 0};
  // 6-arg form on this toolchain (clang-23 / therock-10.0 headers).
  __builtin_amdgcn_tensor_load_to_lds(g0, g1, z4, z4, z8, 0);
}
#endif

// ---------------------------------------------------------------------------
// Kernel 1: one block per graph. Single streaming pass over this graph's node
// rows (segment_ids sorted -> contiguous range found by binary search).
// Online softmax over leaky_relu logits fused with the weighted feature sum:
//   S[b] = sum_n softmax(z)_n * x_n
// node_feats is read from HBM exactly once (~210 MB -> ~9 us roofline floor).
// Chunks are DMA'd into LDS by the Tensor Data Mover, double-buffered, with
// TENSORcnt-based completion (wave 0 drives the DMA; barriers publish LDS).
// ---------------------------------------------------------------------------
__global__ __launch_bounds__(256) void pool_kernel(
    const float* __restrict__ node, const float* __restrict__ gfeat,
    const int* __restrict__ seg, const float* __restrict__ Wl,
    const float* __restrict__ bl, float* __restrict__ S) {
  const int b   = blockIdx.x;
  const int tid = threadIdx.x;
  const int w   = tid >> 5;
  const int l   = tid & 31;

  __shared__ float xsh[2][CH * SS];
  __shared__ float zsh[CH];
  __shared__ float red[8];

  // c1 = relu(g[b]) . W_logit[0:256] + b_logit   (graph-side logit term)
  float gv = gfeat[(size_t)b * FDIM + tid];
  gv = gv > 0.f ? gv : 0.f;
  float part = warp_sum(gv * Wl[tid]);
  if (l == 0) red[w] = part;
  __syncthreads();
  float c1 = bl[0];
#pragma unroll
  for (int i = 0; i < 8; ++i) c1 += red[i];

  // node-side logit weights for this lane (columns l + 32u)
  float wl2[8];
#pragma unroll
  for (int u = 0; u < 8; ++u) wl2[u] = Wl[FDIM + l + 32 * u];

  // segment bounds via binary search (segment_ids sorted ascending)
  int lo = 0, hi = NNODES;
  while (lo < hi) { int mid = (lo + hi) >> 1; if (seg[mid] < b)     lo = mid + 1; else hi = mid; }
  int start = lo;
  lo = start; hi = NNODES;
  while (lo < hi) { int mid = (lo + hi) >> 1; if (seg[mid] < b + 1) lo = mid + 1; else hi = mid; }
  int end = lo;
#ifdef USE_TDM
  start = __builtin_amdgcn_readfirstlane(start);   // force scalar: feeds D# SGPRs
  end   = __builtin_amdgcn_readfirstlane(end);
#endif

  float m_run = -INFINITY, d_run = 0.f, s_f = 0.f;  // thread owns feature column `tid`
  const int nchunks = (end - start + CH - 1) / CH;

#ifdef USE_TDM
  // Prologue: DMA chunk 0 into buffer 0.
  if (w == 0 && nchunks > 0)
    tdm_load_rows((unsigned)(uintptr_t)&xsh[0][0], node + (size_t)start * FDIM,
                  min(CH, end - start));
#endif

  for (int chunk = 0; chunk < nchunks; ++chunk) {
    const int base = start + chunk * CH;
    const int rows = min(CH, end - base);
    const float* xs = &xsh[chunk & 1][0];

#ifdef USE_TDM
    if (w == 0) {
      const int nb = base + CH;
      if (nb < end) {
        // DMA next chunk into the other buffer while this one is consumed.
        tdm_load_rows((unsigned)(uintptr_t)&xsh[(chunk & 1) ^ 1][0],
                      node + (size_t)nb * FDIM, min(CH, end - nb));
        __builtin_amdgcn_s_wait_tensorcnt(1);  // in-order: current chunk done
      } else {
        __builtin_amdgcn_s_wait_tensorcnt(0);
      }
    }
    __syncthreads();   // publish TDM-written LDS to all waves
#else
    // Fallback: cooperative coalesced float4 staging.
    float* xw = &xsh[chunk & 1][0];
    for (int idx = tid; idx < rows * (FDIM / 4); idx += 256) {
      const int r = idx >> 6, c4 = idx & 63;
      const float4 v = *(const float4*)(node + (size_t)(base + r) * FDIM + c4 * 4);
      *(float4*)&xw[r * SS + c4 * 4] = v;
    }
    __syncthreads();
#endif

    // z_i = leaky_relu(c1 + x_i . W_logit[256:512]); wave w handles rows w, w+8, ...
    for (int i = w; i < rows; i += 8) {
      float acc = 0.f;
#pragma unroll
      for (int u = 0; u < 8; ++u) acc += xs[i * SS + l + 32 * u] * wl2[u];
      acc = warp_sum(acc);
      if (l == 0) {
        const float zz = c1 + acc;
        zsh[i] = zz > 0.f ? zz : 0.01f * zz;
      }
    }
    __syncthreads();

    // online softmax update (all threads redundantly track m/d; s_f per-column)
    float cm = -INFINITY;
    for (int i = 0; i < rows; ++i) cm = fmaxf(cm, zsh[i]);
    const float m_new = fmaxf(m_run, cm);
    const float scale = __expf(m_run - m_new);  // exp(-inf)=0 on first chunk
    s_f *= scale;
    d_run *= scale;
    for (int i = 0; i < rows; ++i) {
      const float wgt = __expf(zsh[i] - m_new);
      d_run += wgt;
      s_f += wgt * xs[i * SS + tid];
    }
    m_run = m_new;
    __syncthreads();   // guards zsh and buffer reuse (next-next DMA issues after this)
  }

  S[(size_t)b * FDIM + tid] = (end > start && d_run > 0.f) ? (s_f / d_run) : 0.f;
}

// ---------------------------------------------------------------------------
// Kernel 2: fused tail on [B,256] rows, 16 rows per block, 8 waves.
//   context = elu(S @ W_proj^T + b_proj)       (valid since softmax sums to 1)
//   gi = context @ W_ih^T + b_ih ; gh = g @ W_hh^T + b_hh ; GRU gates -> out
// All GEMMs use V_WMMA_F32_16X16X4_F32 (K=4 per op, 64 steps for K=256).
// ---------------------------------------------------------------------------
__global__ __launch_bounds__(256) void tail_kernel(
    const float* __restrict__ S,   const float* __restrict__ gfeat,
    const float* __restrict__ Wp,  const float* __restrict__ bp,
    const float* __restrict__ Wih, const float* __restrict__ Whh,
    const float* __restrict__ bih, const float* __restrict__ bhh,
    float* __restrict__ out) {
  const int tid = threadIdx.x;
  const int w   = tid >> 5;
  const int l   = tid & 31;
  const int n   = l & 15;        // WMMA N (and A-row M) index for this lane
  const int hi  = l >> 4;        // lane-half selects K pair / row offset
  const int row0 = blockIdx.x * 16;

  __shared__ float Ssh[16 * SS];
  __shared__ float Gsh[16 * SS];
  __shared__ float Csh[16 * SS];

  for (int idx = tid; idx < 16 * (FDIM / 4); idx += 256) {
    const int r = idx >> 6, c4 = idx & 63;
    *(float4*)&Ssh[r * SS + c4 * 4] = *(const float4*)(S     + (size_t)(row0 + r) * FDIM + c4 * 4);
    *(float4*)&Gsh[r * SS + c4 * 4] = *(const float4*)(gfeat + (size_t)(row0 + r) * FDIM + c4 * 4);
  }
  __syncthreads();

  // GEMM1: context = elu(S @ Wp^T + bp). Wave w owns column tiles {2w, 2w+1}.
#pragma unroll
  for (int t = 0; t < 2; ++t) {
    const int ft = 2 * w + t;
    const int f  = ft * 16 + n;          // output column == row of Wp
    v8f acc = {};
    for (int k = 0; k < 64; ++k) {
      const int c = 4 * k + 2 * hi;
      v2f a, bm;
      a.x  = Ssh[n * SS + c];  a.y  = Ssh[n * SS + c + 1];
      bm.x = Wp[(size_t)f * FDIM + c]; bm.y = Wp[(size_t)f * FDIM + c + 1];
      acc = wmma_f32(a, bm, acc);
    }
#pragma unroll
    for (int p = 0; p < 8; ++p) {
      const int m = p + 8 * hi;          // C/D layout: VGPR p -> row p (+8 for hi half)
      float v = acc[p] + bp[f];
      v = v > 0.f ? v : expm1f(v);       // elu
      Csh[m * SS + f] = v;
    }
  }
  __syncthreads();

  // GEMM2+3 + GRU gates. Wave w owns output column tiles {2w, 2w+1}.
#pragma unroll
  for (int t = 0; t < 2; ++t) {
    const int ft = 2 * w + t;
    const int f  = ft * 16 + n;
    v8f ir = {}, iz = {}, in_ = {}, hr = {}, hz = {}, hn = {};
    for (int k = 0; k < 64; ++k) {
      const int c = 4 * k + 2 * hi;
      v2f ac, ag, b0, b1, b2;
      ac.x = Csh[n * SS + c]; ac.y = Csh[n * SS + c + 1];
      ag.x = Gsh[n * SS + c]; ag.y = Gsh[n * SS + c + 1];
      // W_ih rows: [0,256)=r gate, [256,512)=z gate, [512,768)=n gate
      b0.x = Wih[(size_t)(0 * FDIM + f) * FDIM + c]; b0.y = Wih[(size_t)(0 * FDIM + f) * FDIM + c + 1];
      b1.x = Wih[(size_t)(1 * FDIM + f) * FDIM + c]; b1.y = Wih[(size_t)(1 * FDIM + f) * FDIM + c + 1];
      b2.x = Wih[(size_t)(2 * FDIM + f) * FDIM + c]; b2.y = Wih[(size_t)(2 * FDIM + f) * FDIM + c + 1];
      ir  = wmma_f32(ac, b0, ir);
      iz  = wmma_f32(ac, b1, iz);
      in_ = wmma_f32(ac, b2, in_);
      b0.x = Whh[(size_t)(0 * FDIM + f) * FDIM + c]; b0.y = Whh[(size_t)(0 * FDIM + f) * FDIM + c + 1];
      b1.x = Whh[(size_t)(1 * FDIM + f) * FDIM + c]; b1.y = Whh[(size_t)(1 * FDIM + f) * FDIM + c + 1];
      b2.x = Whh[(size_t)(2 * FDIM + f) * FDIM + c]; b2.y = Whh[(size_t)(2 * FDIM + f) * FDIM + c + 1];
      hr = wmma_f32(ag, b0, hr);
      hz = wmma_f32(ag, b1, hz);
      hn = wmma_f32(ag, b2, hn);
    }
    const float bir = bih[f], biz = bih[FDIM + f], bin = bih[2 * FDIM + f];
    const float bhr = bhh[f], bhz = bhh[FDIM + f], bhn = bhh[2 * FDIM + f];
#pragma unroll
    for (int p = 0; p < 8; ++p) {
      const int m = p + 8 * hi;
      const float r_ = 1.f / (1.f + __expf(-((ir[p] + bir) + (hr[p] + bhr))));
      const float z_ = 1.f / (1.f + __expf(-((iz[p] + biz) + (hz[p] + bhz))));
      const float n_ = tanhf((in_[p] + bin) + r_ * (hn[p] + bhn));
      const float g  = Gsh[m * SS + f];
      out[(size_t)(row0 + m) * FDIM + f] = (1.f - z_) * n_ + z_ * g;
    }
  }
}

extern "C" void kernel_launch(void* const* d_in, const int* in_sizes, int n_in,
                              void* d_out, int out_size, void* d_ws, size_t ws_size,
                              hipStream_t stream) {
  const float* node  = (const float*)d_in[0];   // [N,F]
  const float* gfeat = (const float*)d_in[1];   // [B,F]
  const int*   seg   = (const int*)  d_in[2];   // [N]
  const float* Wl    = (const float*)d_in[3];   // [1,2F]
  const float* bl    = (const float*)d_in[4];   // [1]
  const float* Wp    = (const float*)d_in[5];   // [F,F]
  const float* bp    = (const float*)d_in[6];   // [F]
  const float* Wih   = (const float*)d_in[7];   // [3F,F]
  const float* Whh   = (const float*)d_in[8];   // [3F,F]
  const float* bih   = (const float*)d_in[9];   // [3F]
  const float* bhh   = (const float*)d_in[10];  // [3F]

  float* S   = (float*)d_ws;                    // [B,F] attention-weighted feature sums
  float* out = (float*)d_out;                   // [B,F]

  hipLaunchKernelGGL(pool_kernel, dim3(BGRAPH), dim3(256), 0, stream,
                     node, gfeat, seg, Wl, bl, S);
  hipLaunchKernelGGL(tail_kernel, dim3(BGRAPH / 16), dim3(256), 0, stream,
                     S, gfeat, Wp, bp, Wih, Whh, bih, bhh, out);
}